// GLCL_10213432230275
// MI455X (gfx1250) — compile-verified
//
#include <hip/hip_runtime.h>
#include <stdint.h>
#include <math.h>

// ---------------------------------------------------------------------------
// GMSD-style loss, fused stencil + reduction.
// Bandwidth bound: 268 MB in / 4 B out -> ~11.5 us floor at 23.3 TB/s.
// CDNA5 path: async global->LDS tile staging (ASYNCcnt, b128), wave32 blocks.
// Tile layout: LDS col e <-> global x = x0 - 4 + e  (left halo padded to 16B)
//   row = 4 pad | 256 interior | 4 pad  = 264 floats = 66 x 16B chunks.
// ---------------------------------------------------------------------------

#define TX   256          // interior tile width  (== blockDim.x)
#define TY   16           // interior tile height (rows per thread column)
#define ROWS (TY + 2)     // + 1-px halo top/bottom
#define LDSW 264          // full staged row width (floats); 1056 B stride (16B mult.)
#define CH   (LDSW / 4)   // 66 16-byte chunks per row

#if defined(__HIP_DEVICE_COMPILE__) && __has_builtin(__builtin_amdgcn_global_load_async_to_lds_b32)
#define HAS_ASYNC_LDS 1
#else
#define HAS_ASYNC_LDS 0
#endif

#if defined(__HIP_DEVICE_COMPILE__) && __has_builtin(__builtin_amdgcn_global_load_async_to_lds_b128)
#define HAS_ASYNC_B128 1
#else
#define HAS_ASYNC_B128 0
#endif

#if HAS_ASYNC_LDS
typedef int v4i_t __attribute__((__vector_size__(4 * sizeof(int))));
typedef __attribute__((address_space(1))) int   gas_i32;   // global (AS1) scalar
typedef __attribute__((address_space(3))) int   las_i32;   // LDS    (AS3) scalar
typedef __attribute__((address_space(1))) v4i_t gas_v4i;   // global (AS1) int4
typedef __attribute__((address_space(3))) v4i_t las_v4i;   // LDS    (AS3) int4
#endif

__device__ __forceinline__ void stage_b32(const float* __restrict__ gp, float* lp) {
#if HAS_ASYNC_LDS
    // generic LDS ptr low 32 bits are the offset.
    __builtin_amdgcn_global_load_async_to_lds_b32(
        (gas_i32*)(uintptr_t)gp,
        (las_i32*)(uintptr_t)(uint32_t)(uintptr_t)lp,
        0, 0);
#else
    *lp = *gp;
#endif
}

__device__ __forceinline__ void stage_b128(const float* __restrict__ gp, float* lp) {
#if HAS_ASYNC_B128
    __builtin_amdgcn_global_load_async_to_lds_b128(
        (gas_v4i*)(uintptr_t)gp,
        (las_v4i*)(uintptr_t)(uint32_t)(uintptr_t)lp,
        0, 0);
#elif HAS_ASYNC_LDS
    stage_b32(gp + 0, lp + 0);
    stage_b32(gp + 1, lp + 1);
    stage_b32(gp + 2, lp + 2);
    stage_b32(gp + 3, lp + 3);
#else
    *(float4*)lp = *(const float4*)gp;
#endif
}

__device__ __forceinline__ void wait_async_done() {
#if HAS_ASYNC_LDS
#if __has_builtin(__builtin_amdgcn_s_wait_asynccnt)
    __builtin_amdgcn_s_wait_asynccnt(0);
#else
    asm volatile("s_wait_asynccnt 0" ::: "memory");
#endif
#endif
}

__global__ __launch_bounds__(256) void gmsd_tile_kernel(
    const float* __restrict__ im1, const float* __restrict__ im2,
    float* __restrict__ partial, int H, int W)
{
    __shared__ float t1[ROWS][LDSW];
    __shared__ float t2[ROWS][LDSW];
    __shared__ float red[256];

    const int tid = threadIdx.x;
    const int x0  = blockIdx.x * TX;
    const int y0  = blockIdx.y * TY;
    const size_t ibase = (size_t)blockIdx.z * (size_t)H * (size_t)W;

    // ---- stage both image tiles (zero-padded halo) into LDS, 16B chunks ----
    for (int img = 0; img < 2; ++img) {
        const float* src = (img == 0 ? im1 : im2) + ibase;
        float (*tile)[LDSW] = (img == 0 ? t1 : t2);
        for (int i = tid; i < ROWS * CH; i += 256) {
            const int r  = i / CH;            // staged row 0..17
            const int k  = i - r * CH;        // 16B chunk within row
            const int gy = y0 - 1 + r;
            const int gx = x0 - 4 + 4 * k;    // 16B-aligned global x of chunk
            float* lp = &tile[r][4 * k];      // 16B-aligned LDS address
            const bool rowOK = (gy >= 0) && (gy < H);
            if (rowOK && gx >= 0 && gx + 3 < W) {
                stage_b128(src + (size_t)gy * (size_t)W + gx, lp);
            } else if (rowOK && gx + 3 >= 0 && gx < W) {
                // partial chunk (only if W % 4 != 0); per-element fallback
                for (int t = 0; t < 4; ++t) {
                    const int g = gx + t;
                    if (g >= 0 && g < W) stage_b32(src + (size_t)gy * (size_t)W + g, lp + t);
                    else                 lp[t] = 0.0f;
                }
            } else {
                *(float4*)lp = make_float4(0.f, 0.f, 0.f, 0.f);  // zero padding
            }
        }
    }
    wait_async_done();
    __syncthreads();

    // ---- per-thread column walk with rolling 3x3 windows ----
    const int   c     = tid + 4;        // LDS column of this thread's pixel
    const int   px    = x0 + tid;       // global x
    const float TQ    = 0.0026f;
    const float THIRD = (1.0f / 3.0f);
    float lsum = 0.0f;

    float a1 = t1[0][c-1], b1 = t1[0][c], c1 = t1[0][c+1];
    float d1 = t1[1][c-1], e1 = t1[1][c], f1 = t1[1][c+1];
    float a2 = t2[0][c-1], b2 = t2[0][c], c2 = t2[0][c+1];
    float d2 = t2[1][c-1], e2 = t2[1][c], f2 = t2[1][c+1];

    for (int rr = 0; rr < TY; ++rr) {
        const float g1 = t1[rr+2][c-1], h1 = t1[rr+2][c], i1 = t1[rr+2][c+1];
        const float g2 = t2[rr+2][c-1], h2 = t2[rr+2][c], i2 = t2[rr+2][c+1];
        const int py = y0 + rr;
        if (px < W && py < H) {
            // Prewitt/3 gradients (sign irrelevant: magnitude)
            const float gx1 = ((a1 + d1 + g1) - (c1 + f1 + i1)) * THIRD;
            const float gy1 = ((a1 + b1 + c1) - (g1 + h1 + i1)) * THIRD;
            const float gx2 = ((a2 + d2 + g2) - (c2 + f2 + i2)) * THIRD;
            const float gy2 = ((a2 + b2 + c2) - (g2 + h2 + i2)) * THIRD;
            const float m1  = sqrtf(gx1*gx1 + gy1*gy1);
            const float m2  = sqrtf(gx2*gx2 + gy2*gy2);
            const float gms = (2.0f*m1*m2 + TQ) / (m1*m1 + m2*m2 + TQ);
            const float dd  = fabsf(e1 - e2);
            lsum += logf(coshf(dd) + (1.0f - gms));
        }
        // shift windows down one row
        a1 = d1; b1 = e1; c1 = f1;  d1 = g1; e1 = h1; f1 = i1;
        a2 = d2; b2 = e2; c2 = f2;  d2 = g2; e2 = h2; f2 = i2;
    }

    // ---- block tree reduction ----
    red[tid] = lsum;
    __syncthreads();
    for (int s = 128; s > 0; s >>= 1) {
        if (tid < s) red[tid] += red[tid + s];
        __syncthreads();
    }
    if (tid == 0) {
        const int bid = (blockIdx.z * gridDim.y + blockIdx.y) * gridDim.x + blockIdx.x;
        partial[bid] = red[0];
    }
}

__global__ __launch_bounds__(256) void reduce_mean_kernel(
    const float* __restrict__ partial, int n, float* __restrict__ out, double invN)
{
    __shared__ double red[256];
    double s = 0.0;
    for (int i = threadIdx.x; i < n; i += 256) s += (double)partial[i];
    red[threadIdx.x] = s;
    __syncthreads();
    for (int st = 128; st > 0; st >>= 1) {
        if (threadIdx.x < st) red[threadIdx.x] += red[threadIdx.x + st];
        __syncthreads();
    }
    if (threadIdx.x == 0) out[0] = (float)(red[0] * invN);
}

extern "C" void kernel_launch(void* const* d_in, const int* in_sizes, int n_in,
                              void* d_out, int out_size, void* d_ws, size_t ws_size,
                              hipStream_t stream)
{
    const float* im1 = (const float*)d_in[0];
    const float* im2 = (const float*)d_in[1];
    float* out = (float*)d_out;

    const int H = 1024, W = 1024;
    int B = in_sizes[0] / (H * W);
    if (B < 1) B = 1;

    dim3 grid((W + TX - 1) / TX, (H + TY - 1) / TY, B);
    dim3 block(256);
    float* partial = (float*)d_ws;
    const int nPartials = (int)(grid.x * grid.y * grid.z);

    gmsd_tile_kernel<<<grid, block, 0, stream>>>(im1, im2, partial, H, W);

    const double invN = 1.0 / ((double)B * (double)H * (double)W);
    reduce_mean_kernel<<<1, 256, 0, stream>>>(partial, nPartials, out, invN);
}